// GraphAttentionLayer_74431783239689
// MI455X (gfx1250) — compile-verified
//
#include <hip/hip_runtime.h>

typedef float v2f __attribute__((ext_vector_type(2)));
typedef float v8f __attribute__((ext_vector_type(8)));

#define HDIM 64
#define SLEN 200
#define TILES_PER_ROW 13   // ceil(200 / 16)

__global__ __launch_bounds__(256) void gat_kernel(
    const float* __restrict__ emb,    // [VOCAB, 64]
    const float* __restrict__ a,      // [128]: a1 = a[0:64], a2 = a[64:128]
    const int*   __restrict__ seq,    // [B, 200] int32
    const int*   __restrict__ seq_l,  // [B] int32
    float*       __restrict__ out,    // [B, 200, 64]
    int B)
{
    const int lane = threadIdx.x & 31;
    const int wave = threadIdx.x >> 5;
    const int tile = blockIdx.x * 8 + wave;
    if (tile >= B * TILES_PER_ROW) return;

    const int b  = tile / TILES_PER_ROW;
    const int t  = tile - b * TILES_PER_ROW;
    const int s0 = t * 16;
    const int valid = (SLEN - s0) < 16 ? (SLEN - s0) : 16;  // 16 or 8 (tail)

    const int m  = lane & 15;   // row of the 16x* A tile this lane feeds
    const int hi = lane >> 4;   // 0 -> K{0,1}, 1 -> K{2,3}

    // Per-lane position for the tile row (clamped on ragged tail; stores skipped).
    int s = s0 + m; if (s > SLEN - 1) s = SLEN - 1;
    const int sn = (s + 1) % SLEN;           // jnp.roll(seq, -1)

    const long long rowbase = (long long)b * SLEN;
    const int cur = seq[rowbase + s];        // lane m holds idx of position s0+m
    const int nxt = seq[rowbase + sn];
    const int Lb  = seq_l[b];
    const float sl = (float)Lb;
    const int  lim = Lb - 1;

    // sum(a2): lane loads a2[2*lane..2*lane+1], 5-step butterfly reduce.
    v2f a2p = *(const v2f*)(a + HDIM + 2 * lane);
    float suma2 = a2p.x + a2p.y;
    #pragma unroll
    for (int d = 1; d < 32; d <<= 1) suma2 += __shfl_xor(suma2, d, 32);

    // Two accumulators (break WMMA->WMMA RAW chain); fold -sum(a2) into C init.
    // Every B column carries the same a2 chunk, so every C column = d[row].
    v8f c0, c1;
    #pragma unroll
    for (int j = 0; j < 8; ++j) { c0[j] = -suma2; c1[j] = 0.0f; }

    const float* erow = emb + (long long)cur * HDIM;
    const float* nrow = emb + (long long)nxt * HDIM;

    #pragma unroll
    for (int k = 0; k < 16; k += 2) {
        {   // K-chunk k : columns 4k .. 4k+3
            const int col = 4 * k + 2 * hi;
            v2f e2 = *(const v2f*)(erow + col);
            v2f n2 = *(const v2f*)(nrow + col);
            v2f av; av.x = n2.x - e2.x; av.y = n2.y - e2.y;   // diff = en - e
            v2f bv = *(const v2f*)(a + HDIM + col);           // a2 chunk
            c0 = __builtin_amdgcn_wmma_f32_16x16x4_f32(false, av, false, bv,
                                                       (short)0, c0, false, false);
        }
        {   // K-chunk k+1
            const int col = 4 * (k + 1) + 2 * hi;
            v2f e2 = *(const v2f*)(erow + col);
            v2f n2 = *(const v2f*)(nrow + col);
            v2f av; av.x = n2.x - e2.x; av.y = n2.y - e2.y;
            v2f bv = *(const v2f*)(a + HDIM + col);
            c1 = __builtin_amdgcn_wmma_f32_16x16x4_f32(false, av, false, bv,
                                                       (short)0, c1, false, false);
        }
    }

    // d[row]: lane 0 holds rows 0..7 in c[0..7]; lane 16 holds rows 8..15.
    float dval[8];
    #pragma unroll
    for (int j = 0; j < 8; ++j) dval[j] = c0[j] + c1[j];

    // Emit 16 output rows: wave-coalesced 256B stores, float2 per lane.
    #pragma unroll
    for (int mm = 0; mm < 16; ++mm) {
        if (mm >= valid) break;                       // wave-uniform (tail tile)
        const float dm  = __shfl(dval[mm & 7], (mm < 8) ? 0 : 16, 32);
        const float att1 = 1.0f / (1.0f + __expf(-dm));  // softmax[...,1]
        const int  ss = s0 + mm;
        const int  cm = __shfl(cur, mm, 32);
        const int  nm = __shfl(nxt, mm, 32);

        v2f e2 = *(const v2f*)(emb + (long long)cm * HDIM + 2 * lane);
        v2f r;
        if (ss < lim) {                               // wave-uniform branch
            v2f n2 = *(const v2f*)(emb + (long long)nm * HDIM + 2 * lane);
            // attended = e + sl + att1 * (en - e - 1)
            r.x = e2.x + sl + att1 * (n2.x - e2.x - 1.0f);
            r.y = e2.y + sl + att1 * (n2.y - e2.y - 1.0f);
        } else {
            r = e2;                                   // inactive -> raw embedding
        }
        *(v2f*)(out + (rowbase + ss) * HDIM + 2 * lane) = r;
    }
}

extern "C" void kernel_launch(void* const* d_in, const int* in_sizes, int n_in,
                              void* d_out, int out_size, void* d_ws, size_t ws_size,
                              hipStream_t stream) {
    const float* emb   = (const float*)d_in[0];
    const float* a     = (const float*)d_in[1];
    const int*   seq   = (const int*)d_in[2];
    const int*   seq_l = (const int*)d_in[3];
    float*       out   = (float*)d_out;

    const int B     = in_sizes[3];            // 4096
    const int tiles = B * TILES_PER_ROW;      // 13 tiles of 16 positions per row
    const int blocks = (tiles + 7) / 8;       // 8 waves (256 threads) per block

    gat_kernel<<<blocks, 256, 0, stream>>>(emb, a, seq, seq_l, out, B);
}